// FDLModel_49821620633725
// MI455X (gfx1250) — compile-verified
//
#include <hip/hip_runtime.h>

typedef float v2f __attribute__((ext_vector_type(2)));
typedef float v8f __attribute__((ext_vector_type(8)));

#define TWO_PI 6.28318530717958647692f
#define NPIX   262144          // 512*512
#define NPLANES_FWD 51         // 24 x-planes + 27 y-planes
#define CPITCH 513             // LDS column-tile pitch (float2) -> conflict-free

// ---------------- complex helpers (float2) ----------------
__device__ __forceinline__ float2 cadd(float2 a, float2 b){ return make_float2(a.x+b.x, a.y+b.y); }
__device__ __forceinline__ float2 csub(float2 a, float2 b){ return make_float2(a.x-b.x, a.y-b.y); }
__device__ __forceinline__ float2 cmul(float2 a, float2 b){ return make_float2(a.x*b.x - a.y*b.y, a.x*b.y + a.y*b.x); }
// conj(a)*b
__device__ __forceinline__ float2 cmulc(float2 a, float2 b){ return make_float2(a.x*b.x + a.y*b.y, a.x*b.y - a.y*b.x); }

// ---------------- CDNA5 async global<->LDS helpers ----------------
__device__ __forceinline__ void async_load_b64(const void* gaddr, void* lds) {
  unsigned lds_off = (unsigned)(uintptr_t)lds;               // group-segment offset
  unsigned long long ga = (unsigned long long)(uintptr_t)gaddr;
  asm volatile("global_load_async_to_lds_b64 %0, %1, off"
               :: "v"(lds_off), "v"(ga) : "memory");
}
__device__ __forceinline__ void async_store_b64(void* gaddr, const void* lds) {
  unsigned lds_off = (unsigned)(uintptr_t)lds;
  unsigned long long ga = (unsigned long long)(uintptr_t)gaddr;
  asm volatile("global_store_async_from_lds_b64 %1, %0, off"
               :: "v"(lds_off), "v"(ga) : "memory");
}
__device__ __forceinline__ void wait_async0() {
  asm volatile("s_wait_asynccnt 0" ::: "memory");
}

// ---------------- WMMA wrapper: D = A(16x4) * B(4x16) + C ----------------
__device__ __forceinline__ v8f wmma4(v2f a, v2f b, v8f c) {
  return __builtin_amdgcn_wmma_f32_16x16x4_f32(false, a, false, b, (short)0, c, false, false);
}

// ============================================================================
// Wave-level 512-point complex FFT (four-step 16x32) on a per-wave LDS buffer
// buf[512] (input x[n] in natural order, output X[k] in natural order).
// sgn = -1 forward, +1 inverse. 'scale' multiplied into final result.
//
//   X[k1 + 16 k2] = sum_{n2<32} w32^{n2 k2} * w512^{n2 k1}
//                      * sum_{n1<16} x[n2 + 32 n1] * w16^{n1 k1}
//
// Step A (DFT-16) and Step C (DFT-32) run on v_wmma_f32_16x16x4_f32.
// Complex matmul via 4 real WMMAs; subtraction by negating B (Step A) or A (Step C) in VALU.
// A operand layout: lane L, vgpr j -> A[L%16][4c + j + 2*(L/16)]
// B operand layout: lane L, vgpr j -> B[j + 2*(L/16)][L%16]
// C/D layout:       lane L, vgpr r -> D[r + 8*(L/16)][L%16]
// ============================================================================
__device__ void wave_fft512(float2* __restrict__ buf, int lane, float sgn, float scale) {
  const int half = lane >> 4;   // 0 or 1
  const int l16  = lane & 15;

  // ---- DFT-16 basis as A operands, 4 K-chunks of 4 columns each ----
  v2f crA[4], ciA[4];
#pragma unroll
  for (int c = 0; c < 4; ++c) {
#pragma unroll
    for (int j = 0; j < 2; ++j) {
      int n1 = 4*c + j + 2*half;           // column (K index) of D16
      int t  = (l16 * n1) & 15;            // reduced index -> angle in [0,2pi)
      float s, co;
      __sincosf(sgn * (TWO_PI/16.0f) * (float)t, &s, &co);
      crA[c][j] = co;  ciA[c][j] = s;
    }
  }

  // ---- Step A: Y = D16 * M, twiddle, store Z[k1][n2] at buf[k1*32+n2] ----
  // M[n1][n2] = buf[n2 + 32*n1]. Tile t covers n2 in [16t,16t+16).
  // Tile 0's Z writes (addr%32 < 16) are disjoint from tile 1's M reads.
#pragma unroll
  for (int t = 0; t < 2; ++t) {
    v8f yr = {0,0,0,0,0,0,0,0};
    v8f yi = {0,0,0,0,0,0,0,0};
    int n2 = 16*t + l16;
#pragma unroll
    for (int c = 0; c < 4; ++c) {
      float2 m0 = buf[n2 + 32*(4*c + 0 + 2*half)];
      float2 m1 = buf[n2 + 32*(4*c + 1 + 2*half)];
      v2f br  = { m0.x,  m1.x};
      v2f bi  = { m0.y,  m1.y};
      v2f bin = {-m0.y, -m1.y};
      yr = wmma4(crA[c], br,  yr);   // + Cr*Mr
      yr = wmma4(ciA[c], bin, yr);   // - Ci*Mi
      yi = wmma4(crA[c], bi,  yi);   // + Cr*Mi
      yi = wmma4(ciA[c], br,  yi);   // + Ci*Mr
    }
#pragma unroll
    for (int r = 0; r < 8; ++r) {
      int k1 = r + 8*half;
      int tw = (k1 * n2) & 511;
      float s, co;
      __sincosf(sgn * (TWO_PI/512.0f) * (float)tw, &s, &co);
      float zr = yr[r]*co - yi[r]*s;
      float zi = yr[r]*s  + yi[r]*co;
      buf[k1*32 + n2] = make_float2(zr, zi);
    }
  }

  // ---- Step C: X = Z(16x32) * D32(32x32); two k2-tiles, 8 K-chunks ----
  v8f xr0 = {0,0,0,0,0,0,0,0}, xi0 = {0,0,0,0,0,0,0,0};
  v8f xr1 = {0,0,0,0,0,0,0,0}, xi1 = {0,0,0,0,0,0,0,0};
#pragma unroll
  for (int c = 0; c < 8; ++c) {
    // A operand = Z rows (M index = l16), K = n2 chunk
    float2 z0 = buf[l16*32 + 4*c + 0 + 2*half];
    float2 z1 = buf[l16*32 + 4*c + 1 + 2*half];
    v2f azr  = { z0.x,  z1.x};
    v2f azi  = { z0.y,  z1.y};
    v2f azin = {-z0.y, -z1.y};
#pragma unroll
    for (int t = 0; t < 2; ++t) {
      v2f brD, biD;
#pragma unroll
      for (int j = 0; j < 2; ++j) {
        int n2 = 4*c + j + 2*half;         // row of D32 (K index)
        int k2 = 16*t + l16;               // column of D32
        int m  = (n2 * k2) & 31;
        float s, co;
        __sincosf(sgn * (TWO_PI/32.0f) * (float)m, &s, &co);
        brD[j] = co;  biD[j] = s;
      }
      if (t == 0) {
        xr0 = wmma4(azr,  brD, xr0);  xr0 = wmma4(azin, biD, xr0);
        xi0 = wmma4(azr,  biD, xi0);  xi0 = wmma4(azi,  brD, xi0);
      } else {
        xr1 = wmma4(azr,  brD, xr1);  xr1 = wmma4(azin, biD, xr1);
        xi1 = wmma4(azr,  biD, xi1);  xi1 = wmma4(azi,  brD, xi1);
      }
    }
  }
  // Store X[k], k = k1 + 16*k2 (all Z reads already done -> safe to overwrite)
#pragma unroll
  for (int t = 0; t < 2; ++t) {
#pragma unroll
    for (int r = 0; r < 8; ++r) {
      int k = (r + 8*half) + 16*(16*t + l16);
      float re = (t ? xr1[r] : xr0[r]) * scale;
      float im = (t ? xi1[r] : xi0[r]) * scale;
      buf[k] = make_float2(re, im);
    }
  }
}

// ============================================================================
// Kernels. Block = 256 threads = 8 waves. Grids divide exactly -> full blocks,
// uniform control flow, EXEC all ones wherever WMMA executes.
// ============================================================================

__global__ __launch_bounds__(256)
void k_fwd_rows(const float* __restrict__ xin, const float* __restrict__ yin,
                float2* __restrict__ spec) {
  __shared__ float2 lbuf[8*512];
  int wave = threadIdx.x >> 5, lane = threadIdx.x & 31;
  float2* buf = lbuf + wave*512;
  int line  = blockIdx.x*8 + wave;           // over 51*512 lines
  int plane = line >> 9, row = line & 511;
  const float* src = (plane < 24)
      ? (xin + (size_t)plane*NPIX + (size_t)row*512)
      : (yin + (size_t)(plane - 24)*NPIX + (size_t)row*512);
#pragma unroll
  for (int i = 0; i < 16; ++i) { int n = lane + 32*i; buf[n] = make_float2(src[n], 0.0f); }
  wave_fft512(buf, lane, -1.0f, 1.0f);
  float2* dst = spec + (size_t)plane*NPIX + (size_t)row*512;
#pragma unroll
  for (int i = 0; i < 16; ++i) { int n = lane + 32*i; dst[n] = buf[n]; }
}

// Column FFT, in place. Block handles a 512x8 column tile of one plane:
// coalesced 64B row segments, staged via async global<->LDS copies (ASYNCcnt),
// LDS pitch 513 float2 keeps both the tile stores and the FFT accesses
// bank-conflict-free. Wave w transforms column col0+w.
__global__ __launch_bounds__(256)
void k_fft_cols(float2* __restrict__ spec, float sgn, float scale) {
  __shared__ float2 lbuf[8*CPITCH];
  int wave = threadIdx.x >> 5, lane = threadIdx.x & 31;
  int plane = blockIdx.x >> 6;               // 64 blocks per plane
  int col0  = (blockIdx.x & 63) * 8;
  int cl = threadIdx.x & 7;                  // column within tile
  int r0 = threadIdx.x >> 3;                 // row (0..31), step 32
  float2* gbase = spec + (size_t)plane*NPIX + col0 + cl;
#pragma unroll
  for (int i = 0; i < 16; ++i) {
    int r = r0 + 32*i;
    async_load_b64(gbase + (size_t)r*512, &lbuf[cl*CPITCH + r]);
  }
  wait_async0();
  __syncthreads();
  wave_fft512(lbuf + wave*CPITCH, lane, sgn, scale);
  __syncthreads();
#pragma unroll
  for (int i = 0; i < 16; ++i) {
    int r = r0 + 32*i;
    async_store_b64(gbase + (size_t)r*512, &lbuf[cl*CPITCH + r]);
  }
  wait_async0();
}

__global__ __launch_bounds__(256)
void k_inv_rows_real(float2* __restrict__ spec, float* __restrict__ out) {
  __shared__ float2 lbuf[8*512];
  int wave = threadIdx.x >> 5, lane = threadIdx.x & 31;
  float2* buf = lbuf + wave*512;
  int line  = blockIdx.x*8 + wave;           // over 24*512 lines
  int plane = line >> 9, row = line & 511;
  const float2* src = spec + (size_t)plane*NPIX + (size_t)row*512;
#pragma unroll
  for (int i = 0; i < 16; ++i) {
    int n = lane + 32*i;
    async_load_b64(src + n, &buf[n]);
  }
  wait_async0();
  wave_fft512(buf, lane, 1.0f, 1.0f/512.0f);
  float* dst = out + (size_t)plane*NPIX + (size_t)row*512;
#pragma unroll
  for (int i = 0; i < 16; ++i) { int n = lane + 32*i; dst[n] = buf[n].x; }
}

// ----------------------------------------------------------------------------
// Per-bin FDL solve. One thread per frequency bin (p,q).
// M = A^H A + rho I  (8x8 Hermitian), rhs_c = A^H Y_c + rho X_c; Cholesky.
// Writes G in place over the X planes (bin-local, so in-place is safe).
// ----------------------------------------------------------------------------
#define TRI(i,j) ((i)*((i)+1)/2 + (j))

__global__ __launch_bounds__(256)
void k_solve(float2* __restrict__ spec,
             const float* __restrict__ up, const float* __restrict__ vp,
             const float* __restrict__ dp, const float* __restrict__ rp) {
  int tid = blockIdx.x*256 + threadIdx.x;    // 0..262143
  int p = tid >> 9, q = tid & 511;
  float fy = (p < 256) ? (float)p : (float)(p - 512);
  float fx = (q < 256) ? (float)q : (float)(q - 512);
  float wyv = (TWO_PI/512.0f)*fy;
  float wxv = (TWO_PI/512.0f)*fx;
  float rho = rp[0];

  float dd[8];
#pragma unroll
  for (int k = 0; k < 8; ++k) dd[k] = dp[k];

  // lower triangle of A^H A
  float2 L[36];
#pragma unroll
  for (int i = 0; i < 36; ++i) L[i] = make_float2(0.f, 0.f);

  // rhs_c = rho * X_c  (accumulate A^H Y below)
  float2 rhs[3][8];
#pragma unroll
  for (int c = 0; c < 3; ++c)
#pragma unroll
    for (int k = 0; k < 8; ++k) {
      float2 xv = spec[(size_t)(c*8 + k)*NPIX + tid];
      rhs[c][k] = make_float2(rho*xv.x, rho*xv.y);
    }

  for (int vv = 0; vv < 9; ++vv) {
    float shift = up[vv]*wxv + vp[vv]*wyv;
    float2 a[8];
#pragma unroll
    for (int k = 0; k < 8; ++k) {
      float s, co; __sincosf(shift*dd[k], &s, &co);
      a[k] = make_float2(co, s);             // A[v][k] = exp(+i shift d_k)
    }
#pragma unroll
    for (int k = 0; k < 8; ++k)
#pragma unroll
      for (int l = 0; l <= k; ++l)
        L[TRI(k,l)] = cadd(L[TRI(k,l)], cmulc(a[k], a[l]));   // conj(a_k)*a_l
#pragma unroll
    for (int c = 0; c < 3; ++c) {
      float2 yv = spec[(size_t)(24 + c*9 + vv)*NPIX + tid];
#pragma unroll
      for (int k = 0; k < 8; ++k)
        rhs[c][k] = cadd(rhs[c][k], cmulc(a[k], yv));
    }
  }

  // In-place complex Cholesky of M = (lower ATA) + rho I
  float invd[8];
#pragma unroll
  for (int j = 0; j < 8; ++j) {
    float dre = L[TRI(j,j)].x + rho;
#pragma unroll
    for (int k = 0; k < j; ++k) { float2 t = L[TRI(j,k)]; dre -= t.x*t.x + t.y*t.y; }
    float lr  = sqrtf(fmaxf(dre, 1e-30f));
    float inv = 1.0f / lr;
    invd[j] = inv;
    L[TRI(j,j)] = make_float2(lr, 0.f);
#pragma unroll
    for (int i = j + 1; i < 8; ++i) {
      float2 s = L[TRI(i,j)];
#pragma unroll
      for (int k = 0; k < j; ++k)
        s = csub(s, cmulc(L[TRI(j,k)], L[TRI(i,k)]));   // L[i][k]*conj(L[j][k])
      L[TRI(i,j)] = make_float2(s.x*inv, s.y*inv);
    }
  }

  // The reference leaves bins (p==256, 1<=q<256) zero (mirror-range gap).
  bool zero_bin = (p == 256) && (q >= 1) && (q < 256);

#pragma unroll
  for (int c = 0; c < 3; ++c) {
    float2 y[8], x[8];
#pragma unroll
    for (int i = 0; i < 8; ++i) {            // forward: L y = rhs
      float2 s = rhs[c][i];
#pragma unroll
      for (int j = 0; j < i; ++j) s = csub(s, cmul(L[TRI(i,j)], y[j]));
      y[i] = make_float2(s.x*invd[i], s.y*invd[i]);
    }
#pragma unroll
    for (int i = 7; i >= 0; --i) {           // backward: L^H x = y
      float2 s = y[i];
#pragma unroll
      for (int j = i + 1; j < 8; ++j) s = csub(s, cmulc(L[TRI(j,i)], x[j]));
      x[i] = make_float2(s.x*invd[i], s.y*invd[i]);
    }
#pragma unroll
    for (int k = 0; k < 8; ++k)
      spec[(size_t)(c*8 + k)*NPIX + tid] = zero_bin ? make_float2(0.f, 0.f) : x[k];
  }
}

// ============================================================================
extern "C" void kernel_launch(void* const* d_in, const int* in_sizes, int n_in,
                              void* d_out, int out_size, void* d_ws, size_t ws_size,
                              hipStream_t stream) {
  (void)in_sizes; (void)n_in; (void)out_size;
  const float* x   = (const float*)d_in[0];   // [1,3,8,512,512]
  const float* y   = (const float*)d_in[1];   // [1,3,9,512,512]
  const float* u   = (const float*)d_in[2];   // [9]
  const float* v   = (const float*)d_in[3];   // [9]
  const float* d   = (const float*)d_in[4];   // [8]
  const float* rho = (const float*)d_in[5];   // [1]
  float*  out  = (float*)d_out;               // [1,3,8,512,512]
  float2* spec = (float2*)d_ws;               // 51 complex planes (X:0..23, Y:24..50)

  if (ws_size < (size_t)NPLANES_FWD * NPIX * sizeof(float2)) return;

  dim3 blk(256);
  k_fwd_rows     <<<dim3(NPLANES_FWD*512/8), blk, 0, stream>>>(x, y, spec);
  k_fft_cols     <<<dim3(NPLANES_FWD*64),    blk, 0, stream>>>(spec, -1.0f, 1.0f);
  k_solve        <<<dim3(NPIX/256),          blk, 0, stream>>>(spec, u, v, d, rho);
  k_fft_cols     <<<dim3(24*64),             blk, 0, stream>>>(spec,  1.0f, 1.0f/512.0f);
  k_inv_rows_real<<<dim3(24*512/8),          blk, 0, stream>>>(spec, out);
}